// RobinShuffle2d_43250320671165
// MI455X (gfx1250) — compile-verified
//
#include <hip/hip_runtime.h>
#include <stdint.h>

typedef __attribute__((ext_vector_type(2))) float v2f;
typedef __attribute__((ext_vector_type(8))) float v8f;

#define N_IMG 16
#define C_IN  256
#define H_IN  64
#define W_IN  64
#define OC    128
#define HP    63
#define WP    63
#define KDIM  1024            // C_IN * KH * KW
#define KC    64              // K per chunk = 16 channels * 4 taps
#define NCHUNK (KDIM / KC)    // 16
#define PIX_TILE 64           // N tile per block (4 x 16-wide WMMA tiles)
#define LDS_ROW  68           // 64 + 4 pad -> conflict-free ds_load_b64
#define BUF_FLOATS (PIX_TILE * LDS_ROW)   // 4352 floats = 17408 B / buffer

__device__ __forceinline__ v8f wmma4(v2f a, v2f b, v8f c) {
  // D(16x16,f32) = A(16x4,f32) x B(4x16,f32) + C
  return __builtin_amdgcn_wmma_f32_16x16x4_f32(
      false, a, false, b, (short)0, c, false, false);
}

__global__ __launch_bounds__(256)
void robin_shuffle_wmma(const float* __restrict__ x,
                        const float* __restrict__ wgt,
                        const float* __restrict__ bias,
                        float* __restrict__ out)
{
  __shared__ float Bsh[2][BUF_FLOATS];   // 34816 B

  const int cls = blockIdx.y;            // parity class: (h%2)*2 + (w%2)
  const int ph  = cls >> 1, pw = cls & 1;
  const int Hc  = 32 - ph;
  const int Wc  = 32 - pw;
  const int HWc = Hc * Wc;
  const int pcnt = N_IMG * HWc;          // pixels in this class (<= 16384)
  const int pix_base = blockIdx.x * PIX_TILE;
  if (pix_base >= pcnt) return;          // uniform per block

  const int tid  = threadIdx.x;
  const int lane = tid & 31;
  const int wave = tid >> 5;             // 0..7 -> oc slab

  // ---------- loader-side geometry: pixel = pix_base + tid%64 ----------
  int lp  = pix_base + (tid & 63);
  int lpc = (lp < pcnt) ? lp : (pcnt - 1);       // clamp; garbage never stored
  int img = lpc / HWc;
  int rem = lpc - img * HWc;
  int ih  = rem / Wc;
  int iw  = rem - ih * Wc;
  int hh  = ph + 2 * ih;                         // top-left of 2x2 window
  int ww  = pw + 2 * iw;
  const float* xpix = x + ((size_t)img * C_IN) * (H_IN * W_IN)
                        + hh * W_IN + ww;
  const int cgrp = tid >> 6;                     // 0..3 -> 4 channels each
  const unsigned lds_base = (unsigned)(uintptr_t)(void*)&Bsh[0][0];
  const unsigned lrow = lds_base + (unsigned)(tid & 63) * (LDS_ROW * 4);

  // DMA one K-chunk of the im2col B tile straight into LDS (ASYNCcnt path).
  // Per channel c: taps (0,0),(0,1) share one addr (offset 0/4), likewise
  // (1,0),(1,1) -- instruction offset applies to both global and LDS sides.
  auto stage_async = [&](int ch, int buf) {
    unsigned lbase = lrow + (unsigned)buf * (BUF_FLOATS * 4);
#pragma unroll
    for (int u = 0; u < 4; ++u) {
      int cc = cgrp * 4 + u;                     // channel within chunk, 0..15
      const void* g0 = (const void*)(xpix + (size_t)(ch * 16 + cc) * (H_IN * W_IN));
      const void* g1 = (const void*)((const float*)g0 + W_IN);
      unsigned l0 = lbase + (unsigned)cc * 16u;  // kk = cc*4 .. cc*4+3
      unsigned l1 = l0 + 8u;
      asm volatile(
        "global_load_async_to_lds_b32 %0, %2, off\n\t"
        "global_load_async_to_lds_b32 %0, %2, off offset:4\n\t"
        "global_load_async_to_lds_b32 %1, %3, off\n\t"
        "global_load_async_to_lds_b32 %1, %3, off offset:4"
        :: "v"(l0), "v"(l1), "v"(g0), "v"(g1)
        : "memory");
    }
  };

  // ---------- compute-side setup ----------
  const int mrow = lane & 15;                    // M index within 16
  const int koff = (lane < 16) ? 0 : 2;          // A/B half-wave K offset
  const int ocm  = wave * 16 + mrow;
  const float* arow  = wgt + (size_t)(ocm * 4 + cls) * KDIM + koff;
  const float* bbase = &Bsh[0][mrow * LDS_ROW + koff];

  v8f acc0 = {}, acc1 = {}, acc2 = {}, acc3 = {};

  stage_async(0, 0);
  asm volatile("s_wait_asynccnt 0x0" ::: "memory");
  __syncthreads();

  for (int ch = 0; ch < NCHUNK; ++ch) {
    const int buf = ch & 1;
    if (ch + 1 < NCHUNK) stage_async(ch + 1, buf ^ 1);  // DMA next chunk

    const float* ap = arow + ch * KC;
    const float* bp = bbase + buf * BUF_FLOATS;

    // software-pipelined fragment loads: step s+1 in flight during step s WMMAs
    v2f a  = *(const v2f*)(ap);
    v2f b0 = *(const v2f*)(bp);
    v2f b1 = *(const v2f*)(bp + 16 * LDS_ROW);
    v2f b2 = *(const v2f*)(bp + 32 * LDS_ROW);
    v2f b3 = *(const v2f*)(bp + 48 * LDS_ROW);
#pragma unroll
    for (int s = 0; s < KC / 4; ++s) {
      const int sn = (s + 1 < KC / 4) ? (s + 1) : s;   // clamp: uniform flow
      v2f na  = *(const v2f*)(ap + sn * 4);                  // global_load_b64
      v2f nb0 = *(const v2f*)(bp + sn * 4);                  // ds_load_b64
      v2f nb1 = *(const v2f*)(bp + 16 * LDS_ROW + sn * 4);
      v2f nb2 = *(const v2f*)(bp + 32 * LDS_ROW + sn * 4);
      v2f nb3 = *(const v2f*)(bp + 48 * LDS_ROW + sn * 4);
      acc0 = wmma4(a, b0, acc0);
      acc1 = wmma4(a, b1, acc1);
      acc2 = wmma4(a, b2, acc2);
      acc3 = wmma4(a, b3, acc3);
      a = na; b0 = nb0; b1 = nb1; b2 = nb2; b3 = nb3;
    }

    if (ch + 1 < NCHUNK)
      asm volatile("s_wait_asynccnt 0x0" ::: "memory");   // my DMAs landed
    __syncthreads();                                      // everyone's landed
  }

  // ---------- epilogue: bias + scattered stores ----------
  // acc VGPR r holds M=r (lanes 0-15) / M=r+8 (lanes 16-31), N = lane%16
  const int oc_base = wave * 16 + ((lane >> 4) << 3);
#pragma unroll
  for (int t = 0; t < 4; ++t) {
    int p = pix_base + t * 16 + (lane & 15);
    if (p < pcnt) {
      int im = p / HWc;
      int rm = p - im * HWc;
      int oi = rm / Wc;
      int oj = rm - oi * Wc;
      int oh = ph + 2 * oi;
      int ow = pw + 2 * oj;
      float* op = out + ((size_t)im * OC) * (HP * WP) + oh * WP + ow;
      v8f acc = (t == 0) ? acc0 : (t == 1) ? acc1 : (t == 2) ? acc2 : acc3;
#pragma unroll
      for (int r = 0; r < 8; ++r) {
        int oc = oc_base + r;
        op[(size_t)oc * (HP * WP)] = acc[r] + bias[oc * 4 + cls];
      }
    }
  }
}

extern "C" void kernel_launch(void* const* d_in, const int* in_sizes, int n_in,
                              void* d_out, int out_size, void* d_ws, size_t ws_size,
                              hipStream_t stream) {
  const float* x   = (const float*)d_in[0];
  const float* w   = (const float*)d_in[1];
  const float* b   = (const float*)d_in[2];
  float* out = (float*)d_out;
  // max pixels per class = 16*32*32 = 16384 -> 256 tiles of 64; 4 classes
  dim3 grid(256, 4, 1);
  robin_shuffle_wmma<<<grid, 256, 0, stream>>>(x, w, b, out);
}